// VectorQuantizer_88562225643603
// MI455X (gfx1250) — compile-verified
//
#include <hip/hip_runtime.h>

typedef __attribute__((ext_vector_type(2))) float v2f;
typedef __attribute__((ext_vector_type(8))) float v8f;
typedef unsigned int u32x4 __attribute__((ext_vector_type(4)));
typedef int          i32x8 __attribute__((ext_vector_type(8)));
typedef int          i32x4 __attribute__((ext_vector_type(4)));

#define DIMX 1024           // feature dim
#define KC   8192           // codebook size
#define ROWS_PER_WG 32      // two 16-row WMMA tiles per workgroup
#define DBLK 64             // dims staged per codebook tile (64 KB DMA per stage)
#define NDBLK (DIMX / DBLK)             // 16
#define CODES_PER_BLK 256   // 8 waves * 32 codes
#define NCBLK (KC / CODES_PER_BLK)      // 32
#define CB_ROW 272          // padded LDS row stride (floats) -> conflict-free banks
#define NSTAGE (NCBLK * NDBLK)          // 512

#if __has_builtin(__builtin_amdgcn_tensor_load_to_lds) && \
    __has_builtin(__builtin_amdgcn_s_wait_tensorcnt)
#define USE_TDM 1
#else
#define USE_TDM 0
#endif

#if USE_TDM
// One TDM descriptor stages a [DBLK dims x 256 codes] f32 tile from the
// codebook (row stride KC) into LDS, padding 16 dwords after each 256-dword
// row so the LDS tile has a 272-float row stride (bank-conflict-free).
__device__ __forceinline__ void tdm_stage(const float* gsrc, unsigned lds_addr)
{
    unsigned long long ga = (unsigned long long)(size_t)gsrc;
    u32x4 g0 = {
        1u,                                                    // count=1 (valid user D#)
        lds_addr,                                              // LDS byte address
        (unsigned)(ga & 0xFFFFFFFFu),                          // global_addr[31:0]
        (unsigned)((ga >> 32) & 0x1FFFFFFu) | (2u << 30)       // global_addr[56:32], type=2
    };
    i32x8 g1 = {
        (2 << 16) | (1 << 20) | (7 << 22) | (15 << 25),        // data_size=4B, pad_en,
                                                               // interval=256dw, amount=16dw
        (int)(8192u << 16),                                    // tensor_dim0 = 8192 (lo16)
        (int)(1024u << 16),                                    // dim0 hi=0, tensor_dim1=1024
        (int)(256u  << 16),                                    // dim1 hi=0, tile_dim0=256
        DBLK,                                                  // tile_dim1=64, tile_dim2=0
        KC,                                                    // tensor_dim0_stride = 8192
        0, 0                                                   // stride hi / dim1_stride
    };
    i32x4 zg4 = { 0, 0, 0, 0 };                                // group2 (unused, 2-D tile)
    i32x4 zg3 = { 0, 0, 0, 0 };                                // group3 (unused)
    i32x8 zg8 = { 0, 0, 0, 0, 0, 0, 0, 0 };                    // extra group (clang-23 form)
    __builtin_amdgcn_tensor_load_to_lds(g0, g1, zg4, zg3, zg8, 0);
}
#endif

// ---------------------------------------------------------------------------
// Pass 1: fused f32 GEMM + row argmin.
// WG = 8 waves, 32 rows of z vs all 8192 codes. Each wave: 2 row-tiles x
// 2 code-tiles = 4 independent v_wmma_f32_16x16x4_f32 chains; A/B fragments
// each feed two WMMAs (256 B LDS per WMMA). Codebook tiles arrive via the
// Tensor Data Mover, double buffered; one barrier per 64-KB staged tile.
// ---------------------------------------------------------------------------
__launch_bounds__(256, 1)
__global__ void vq_argmin_kernel(const float* __restrict__ z,
                                 const float* __restrict__ cb,
                                 int* __restrict__ idxOut)
{
    __shared__ float zf[2 * 512 * 16 * 2];      // 128 KB: two pair-swizzled 16-row regions
    __shared__ float cf[2][DBLK * CB_ROW];      // 2 x 68 KB double-buffered cb tile
    __shared__ float resv[8][32];
    __shared__ int   resi[8][32];

    const int t    = threadIdx.x;
    const int lane = t & 31;
    const int wv   = t >> 5;        // wave 0..7
    const int half = lane >> 4;
    const int ln16 = lane & 15;
    const long rowBase = (long)blockIdx.x * ROWS_PER_WG;

    // ---- stage z tile (32 rows x 1024 dims), coalesced reads, pair layout ----
    {
        const float* zsrc = z + rowBase * DIMX;
        #pragma unroll
        for (int i = 0; i < 128; ++i) {
            int e   = t + i * 256;          // e = row*1024 + d
            int row = e >> 10;
            int d   = e & 1023;
            zf[(row >> 4) * 16384 + (((d >> 1) * 16 + (row & 15)) << 1) + (d & 1)] = zsrc[e];
        }
    }

    float m0[8], m1[8];
    int   mi0[8], mi1[8];
    #pragma unroll
    for (int r = 0; r < 8; ++r) { m0[r] = m1[r] = 3.4e38f; mi0[r] = mi1[r] = 0; }

    v8f acc00 = {}, acc01 = {}, acc10 = {}, acc11 = {};

#if USE_TDM
    if (wv == 0) tdm_stage(cb, (unsigned)(size_t)&cf[0][0]);   // prologue: stage 0
#endif

    for (int s = 0; s < NSTAGE; ++s) {
        const int cblk  = s / NDBLK;
        const int dblk  = s % NDBLK;
        const int buf   = s & 1;
        const int cbase = cblk * CODES_PER_BLK;

#if USE_TDM
        if (wv == 0) __builtin_amdgcn_s_wait_tensorcnt(0);     // stage s landed
        __syncthreads();                                       // visible to all; WAR cleared
        if (wv == 0 && s + 1 < NSTAGE) {                       // prefetch stage s+1
            const int nc = (s + 1) / NDBLK, nd = (s + 1) % NDBLK;
            tdm_stage(cb + (long)(nd * DBLK) * KC + nc * CODES_PER_BLK,
                      (unsigned)(size_t)&cf[buf ^ 1][0]);
        }
#else
        __syncthreads();
        {   // manual fallback staging of cf[buf]
            const float* src = cb + (long)(dblk * DBLK) * KC + cbase;
            #pragma unroll
            for (int i = 0; i < DBLK; ++i) {
                int e = t + i * 256;
                int j = e >> 8;
                int c = e & 255;
                cf[buf][j * CB_ROW + c] = src[(long)j * KC + c];
            }
        }
        __syncthreads();
#endif

        if (dblk == 0) { acc00 = (v8f){}; acc01 = (v8f){}; acc10 = (v8f){}; acc11 = (v8f){}; }

        const float* cfb = cf[buf];
        const int c0 = wv * 32 + ln16;      // code column within 256-block (tile 0)
        #pragma unroll
        for (int kk = 0; kk < DBLK / 4; ++kk) {
            const int d0 = dblk * DBLK + kk * 4;
            const int pA = (d0 >> 1) + half;
            v2f a0 = *(const v2f*)&zf[        ((pA * 16 + ln16) << 1)];
            v2f a1 = *(const v2f*)&zf[16384 + ((pA * 16 + ln16) << 1)];
            const int j = kk * 4 + half * 2;            // dim row within tile
            v2f b0, b1;
            b0.x = cfb[ j      * CB_ROW + c0];
            b0.y = cfb[(j + 1) * CB_ROW + c0];
            b1.x = cfb[ j      * CB_ROW + c0 + 16];
            b1.y = cfb[(j + 1) * CB_ROW + c0 + 16];
            acc00 = __builtin_amdgcn_wmma_f32_16x16x4_f32(false, a0, false, b0, (short)0, acc00, false, false);
            acc01 = __builtin_amdgcn_wmma_f32_16x16x4_f32(false, a0, false, b1, (short)0, acc01, false, false);
            acc10 = __builtin_amdgcn_wmma_f32_16x16x4_f32(false, a1, false, b0, (short)0, acc10, false, false);
            acc11 = __builtin_amdgcn_wmma_f32_16x16x4_f32(false, a1, false, b1, (short)0, acc11, false, false);
        }

        if (dblk == NDBLK - 1) {            // full 1024-dim accumulated: argmin update
            const int ca  = cbase + wv * 32 + ln16;
            const int cbn = ca + 16;
            #pragma unroll
            for (int r = 0; r < 8; ++r) {
                float v; bool lt;
                v = acc00[r];
                lt = (v < m0[r]) || (v == m0[r] && ca < mi0[r]);
                m0[r] = lt ? v : m0[r];  mi0[r] = lt ? ca : mi0[r];
                v = acc01[r];
                lt = (v < m0[r]) || (v == m0[r] && cbn < mi0[r]);
                m0[r] = lt ? v : m0[r];  mi0[r] = lt ? cbn : mi0[r];
                v = acc10[r];
                lt = (v < m1[r]) || (v == m1[r] && ca < mi1[r]);
                m1[r] = lt ? v : m1[r];  mi1[r] = lt ? ca : mi1[r];
                v = acc11[r];
                lt = (v < m1[r]) || (v == m1[r] && cbn < mi1[r]);
                m1[r] = lt ? v : m1[r];  mi1[r] = lt ? cbn : mi1[r];
            }
        }
    }

    // ---- reduce across the 16 lanes sharing each row (wave32 shuffles) ----
    #pragma unroll
    for (int r = 0; r < 8; ++r) {
        #pragma unroll
        for (int off = 8; off >= 1; off >>= 1) {
            float ov; int oi; bool take;
            ov = __shfl_xor(m0[r], off, 32); oi = __shfl_xor(mi0[r], off, 32);
            take = (ov < m0[r]) || (ov == m0[r] && oi < mi0[r]);
            m0[r] = take ? ov : m0[r];  mi0[r] = take ? oi : mi0[r];
            ov = __shfl_xor(m1[r], off, 32); oi = __shfl_xor(mi1[r], off, 32);
            take = (ov < m1[r]) || (ov == m1[r] && oi < mi1[r]);
            m1[r] = take ? ov : m1[r];  mi1[r] = take ? oi : mi1[r];
        }
    }
    if (ln16 == 0) {
        #pragma unroll
        for (int r = 0; r < 8; ++r) {
            resv[wv][     half * 8 + r] = m0[r];
            resi[wv][     half * 8 + r] = mi0[r];
            resv[wv][16 + half * 8 + r] = m1[r];
            resi[wv][16 + half * 8 + r] = mi1[r];
        }
    }
    __syncthreads();
    if (t < 32) {       // reduce across the 8 waves, one thread per row
        float bv = resv[0][t];
        int   bi = resi[0][t];
        #pragma unroll
        for (int w2 = 1; w2 < 8; ++w2) {
            float v  = resv[w2][t];
            int   i2 = resi[w2][t];
            bool take = (v < bv) || (v == bv && i2 < bi);
            bv = take ? v  : bv;
            bi = take ? i2 : bi;
        }
        idxOut[rowBase + t] = bi;
    }
}

// ---------------------------------------------------------------------------
// Pass 2: gather codebook columns. Writes coalesced; column reads stay in the
// 192 MB L2 (codebook is 32 MB).
// ---------------------------------------------------------------------------
__global__ void vq_gather_kernel(const float* __restrict__ cb,
                                 const int* __restrict__ idx,
                                 float* __restrict__ out)
{
    const long n    = blockIdx.x;
    const int  code = idx[n];
    const float* src = cb + code;           // column, stride KC
    float* dst = out + n * DIMX;
    for (int d = threadIdx.x; d < DIMX; d += blockDim.x)
        dst[d] = src[(long)d * KC];
}

extern "C" void kernel_launch(void* const* d_in, const int* in_sizes, int n_in,
                              void* d_out, int out_size, void* d_ws, size_t ws_size,
                              hipStream_t stream) {
    const float* z  = (const float*)d_in[0];   // (16,4096,1024) f32
    const float* cb = (const float*)d_in[1];   // (1024,8192)   f32
    float* out = (float*)d_out;
    int*   idx = (int*)d_ws;                   // 65536 ints = 256 KB scratch

    const int nrows = in_sizes[0] / DIMX;      // 65536
    vq_argmin_kernel<<<nrows / ROWS_PER_WG, 256, 0, stream>>>(z, cb, idx);
    vq_gather_kernel<<<nrows, 256, 0, stream>>>(cb, idx, out);
}